// GeometricNaiveBayes_90958817394872
// MI455X (gfx1250) — compile-verified
//
#include <hip/hip_runtime.h>
#include <hip/hip_bf16.h>

#define B_DIM 256
#define D_DIM 256
#define C_CLS 1000
#define NT    16            // D_DIM/16 tiles along the e (N) dimension
#define LOG_2PI 1.8378770664093453f

// LDS panel staging geometry: each 32-row panel is stored as two 128-column
// half-blocks so the LDS row stride (136 dwords) fits ds_load_2addr's 255-dword
// offset field -> {G[d,e], G[d+1,e]} loads as one instruction, no repack.
#define PANEL_ROWS  32                    // K-rows of G per panel
#define NPANELS     (D_DIM / PANEL_ROWS)  // 8
#define KSTEPS      (PANEL_ROWS / 4)      // 8 WMMA k-steps per panel
#define HALF_COLS   128                   // columns per half-block
#define HALF_STRIDE (HALF_COLS + 8)       // 136 dwords: 8-dword TDM pad per row
#define HALF_DW     (PANEL_ROWS * HALF_STRIDE) // 4352 dwords per half-block
#define SLICE_ROWS  4                     // rows each wave DMAs per panel half

typedef __attribute__((ext_vector_type(2))) float v2f;
typedef __attribute__((ext_vector_type(8))) float v8f;
typedef __attribute__((ext_vector_type(4))) unsigned int u32x4;
typedef __attribute__((ext_vector_type(8))) int i32x8;
typedef __attribute__((ext_vector_type(4))) int i32x4;

__device__ __forceinline__ float half16_sum(float v) {
    v += __shfl_xor(v, 1, 32);
    v += __shfl_xor(v, 2, 32);
    v += __shfl_xor(v, 4, 32);
    v += __shfl_xor(v, 8, 32);
    return v;
}

// TDM: DMA a SLICE_ROWS x 128 fp32 tile (global row stride 256 floats) into
// LDS at lds_off, padding 8 dwords after every 128 so LDS stride = HALF_STRIDE.
__device__ __forceinline__ void tdm_load_half(const float* gsrc, unsigned lds_off) {
    unsigned long long ga = (unsigned long long)(uintptr_t)gsrc;

    u32x4 g0;
    g0[0] = 1u;                                   // count=1, user descriptor
    g0[1] = lds_off;                              // LDS byte address
    g0[2] = (unsigned)(ga & 0xFFFFFFFFu);         // global_addr[31:0]
    g0[3] = (unsigned)((ga >> 32) & 0x1FFFFFFu)   // global_addr[56:32]
          | (2u << 30);                           // type = 2 ("image")

    i32x8 g1;
    g1[0] = (int)((2u << 16)      // data_size = 4 bytes
                | (1u << 20)      // pad_enable
                | (6u << 22)      // pad_interval: 128 dwords
                | (7u << 25));    // pad_amount:   8 dwords
    g1[1] = (int)((unsigned)HALF_COLS << 16);   // tensor_dim0 = 128 (low 16)
    g1[2] = (int)((unsigned)SLICE_ROWS << 16);  // dim0 hi=0; tensor_dim1 = 4
    g1[3] = (int)((unsigned)HALF_COLS << 16);   // dim1 hi=0; tile_dim0 = 128
    g1[4] = SLICE_ROWS;           // tile_dim1 = 4 rows; tile_dim2 = 0
    g1[5] = 256;                  // tensor_dim0_stride = 256 elements (global)
    g1[6] = 0;                    // stride0 hi = 0; tensor_dim1_stride lo = 0
    g1[7] = 0;

    i32x4 z4 = {0, 0, 0, 0};                      // groups 2/3 unused (2D tensor)
    i32x8 z8 = {0, 0, 0, 0, 0, 0, 0, 0};          // trailing words (count=1)
    __builtin_amdgcn_tensor_load_to_lds(g0, g1, z4, z4, z8, 0);
}

// bias[c] = log_priors[c] - 0.5*(sum_d class_log_vars[c,d] + D*log(2pi))
__global__ __launch_bounds__(32) void gnb_bias_kernel(
    const float* __restrict__ log_priors,
    const float* __restrict__ clv,
    float* __restrict__ bias)
{
    const int c    = blockIdx.x;
    const int lane = threadIdx.x;
    float s = 0.0f;
    for (int d = lane; d < D_DIM; d += 32)
        s += clv[(size_t)c * D_DIM + d];
    s += __shfl_xor(s, 1, 32);
    s += __shfl_xor(s, 2, 32);
    s += __shfl_xor(s, 4, 32);
    s += __shfl_xor(s, 8, 32);
    s += __shfl_xor(s, 16, 32);
    if (lane == 0)
        bias[c] = log_priors[c] - 0.5f * (s + (float)D_DIM * LOG_2PI);
}

// 8 waves per block share one batch b; wave w owns 16-class tile
// ct = blockIdx.y*8 + w. G_b is staged through LDS by the Tensor Data Mover in
// 32-row double-buffered panels (2 TDM half-blocks each, TENSORcnt pipelined).
// A (diff) and B fragments are software-pipelined in registers so LDS/global
// latency overlaps the 16 V_WMMA_F32_16X16X4_F32 of each k-step.
__global__ __launch_bounds__(256) void gnb_maha_kernel(
    const float* __restrict__ x,
    const float* __restrict__ G,
    const float* __restrict__ mu,
    const float* __restrict__ bias,
    float* __restrict__ out)
{
    __shared__ float sG[2][2][HALF_DW];   // [buffer][half][rows*HALF_STRIDE]

    const int b     = blockIdx.x;
    const int w     = threadIdx.x >> 5;   // wave id 0..7
    const int lane  = threadIdx.x & 31;
    const int l15   = lane & 15;
    const int hi    = lane >> 4;          // 0: lanes 0-15, 1: lanes 16-31
    const int ct    = blockIdx.y * 8 + w; // class tile 0..63 (63 = padding tile)
    const int cbase = ct * 16;

    const float* __restrict__ Gb = G + (size_t)b * D_DIM * D_DIM;
    const float* __restrict__ xb = x + (size_t)b * D_DIM;

    const v8f zero = {0.f, 0.f, 0.f, 0.f, 0.f, 0.f, 0.f, 0.f};
    v8f acc[NT];
#pragma unroll
    for (int t = 0; t < NT; ++t) acc[t] = zero;

    const int ca  = cbase + l15;
    const int cac = (ca < C_CLS) ? ca : (C_CLS - 1);
    const float* __restrict__ mua = mu + (size_t)cac * D_DIM;

    auto issue = [&](int p) {
        const float* src =
            Gb + (size_t)(p * PANEL_ROWS + w * SLICE_ROWS) * D_DIM;
#pragma unroll
        for (int h = 0; h < 2; ++h) {
            unsigned off = (unsigned)(uintptr_t)
                &sG[p & 1][h][w * SLICE_ROWS * HALF_STRIDE];
            tdm_load_half(src + h * HALF_COLS, off);
        }
    };

    issue(0);
    issue(1);

    for (int p = 0; p < NPANELS; ++p) {
        // 2 TDM ops per panel; in-order -> panel p done when <= 2 outstanding.
        if (p + 1 < NPANELS) __builtin_amdgcn_s_wait_tensorcnt(2);
        else                 __builtin_amdgcn_s_wait_tensorcnt(0);
        __syncthreads();                      // whole panel p visible in LDS

        const float* sp = &sG[p & 1][0][0];

        // Software-pipelined fragment buffers (registers).
        v2f abuf[2];
        v2f bmbuf[2][NT];

        auto loadB = [&](int ks, v2f* dst) {
            const int rr = ks * 4 + 2 * hi;   // panel-local row for this half
#pragma unroll
            for (int t = 0; t < NT; ++t) {
                const int hblk = t >> 3;              // 0: e<128, 1: e>=128
                const int el   = (t & 7) * 16 + l15;  // column within half
                const int base = hblk * HALF_DW + rr * HALF_STRIDE + el;
                v2f bm;
                bm.x = sp[base];                // G[b, d,   e]
                bm.y = sp[base + HALF_STRIDE];  // G[b, d+1, e]  (one 2addr load)
                dst[t] = bm;
            }
        };
        auto loadA = [&](int ks, v2f* dst) {
            const int d = p * PANEL_ROWS + ks * 4 + 2 * hi;
            v2f a;
            a.x = xb[d]     - mua[d];
            a.y = xb[d + 1] - mua[d + 1];
            *dst = a;
        };

        loadA(0, &abuf[0]);
        loadB(0, bmbuf[0]);

#pragma unroll
        for (int ks = 0; ks < KSTEPS; ++ks) {
            const int cur = ks & 1;
            const int nxt = cur ^ 1;
            if (ks + 1 < KSTEPS) {            // prefetch while WMMAs run
                loadA(ks + 1, &abuf[nxt]);
                loadB(ks + 1, bmbuf[nxt]);
            }
#pragma unroll
            for (int t = 0; t < NT; ++t) {
                acc[t] = __builtin_amdgcn_wmma_f32_16x16x4_f32(
                    false, abuf[cur], false, bmbuf[cur][t], (short)0,
                    acc[t], false, false);
            }
        }

        __syncthreads();                      // everyone done reading buf p&1
        if (p + 2 < NPANELS) issue(p + 2);    // overwrite freed buffer
    }

    // Epilogue: C/D layout: VGPR r -> M=r (lanes 0-15) / M=8+r (lanes 16-31),
    // N = l&15. maha[c] = sum_e Y[c,e] * diff[c,e].
    float part[8];
#pragma unroll
    for (int r = 0; r < 8; ++r) part[r] = 0.0f;

    const int crow = cbase + hi * 8;
#pragma unroll
    for (int t = 0; t < NT; ++t) {
        const int e    = t * 16 + l15;
        const float xe = xb[e];
#pragma unroll
        for (int r = 0; r < 8; ++r) {
            const int c  = crow + r;
            const int cc = (c < C_CLS) ? c : (C_CLS - 1);
            part[r] += acc[t][r] * (xe - mu[(size_t)cc * D_DIM + e]);
        }
    }

#pragma unroll
    for (int r = 0; r < 8; ++r) {
        const float m = half16_sum(part[r]);
        const int c = crow + r;
        if (l15 == 0 && c < C_CLS)
            out[(size_t)b * C_CLS + c] = bias[c] - 0.5f * m;
    }
}

// In-place row log_softmax over C=1000 classes; one 256-thread block per row.
__global__ __launch_bounds__(256) void gnb_logsoftmax_kernel(float* __restrict__ out)
{
    const int b   = blockIdx.x;
    const int tid = threadIdx.x;
    __shared__ float red[256];

    float vals[4];
    int n = 0;
    for (int c = tid; c < C_CLS; c += 256) vals[n++] = out[(size_t)b * C_CLS + c];

    float m = -3.402823466e38f;
    for (int i = 0; i < n; ++i) m = fmaxf(m, vals[i]);
    red[tid] = m;
    __syncthreads();
    for (int s = 128; s > 0; s >>= 1) {
        if (tid < s) red[tid] = fmaxf(red[tid], red[tid + s]);
        __syncthreads();
    }
    const float mx = red[0];
    __syncthreads();

    float ssum = 0.0f;
    for (int i = 0; i < n; ++i) ssum += expf(vals[i] - mx);
    red[tid] = ssum;
    __syncthreads();
    for (int s = 128; s > 0; s >>= 1) {
        if (tid < s) red[tid] += red[tid + s];
        __syncthreads();
    }
    const float lse = mx + logf(red[0]);

    n = 0;
    for (int c = tid; c < C_CLS; c += 256) out[(size_t)b * C_CLS + c] = vals[n++] - lse;
}

extern "C" void kernel_launch(void* const* d_in, const int* in_sizes, int n_in,
                              void* d_out, int out_size, void* d_ws, size_t ws_size,
                              hipStream_t stream) {
    const float* x   = (const float*)d_in[0];   // [B, D]
    const float* G   = (const float*)d_in[1];   // [B, D, D]
    const float* lpr = (const float*)d_in[2];   // [C]
    const float* mu  = (const float*)d_in[3];   // [C, D]
    const float* clv = (const float*)d_in[4];   // [C, D]
    float* out  = (float*)d_out;                // [B, C]
    float* bias = (float*)d_ws;                 // C floats of scratch

    gnb_bias_kernel<<<C_CLS, 32, 0, stream>>>(lpr, clv, bias);

    dim3 grid(B_DIM, 8);                        // 8 waves/block * 8 = 64 tiles
    gnb_maha_kernel<<<grid, 256, 0, stream>>>(x, G, mu, bias, out);

    gnb_logsoftmax_kernel<<<B_DIM, 256, 0, stream>>>(out);
}